// SimpleSubgraphEncoder_68856915690053
// MI455X (gfx1250) — compile-verified
//
#include <hip/hip_runtime.h>

#define NN   100000   // nodes
#define NE   1600000  // edges
#define HID  128
#define NG   512      // graphs
#define NLAY 3

typedef float v2f __attribute__((ext_vector_type(2)));
typedef float v8f __attribute__((ext_vector_type(8)));

// out[M x 128] = act( (Xa (+ Xb)) @ W[128x128] (+ bias) ), W row-major [k][n]
// LDS holds W pair-interleaved: sW[kp*256 + c*2 + r] = W[2*kp + r][c]
// so a B fragment {W[k][c], W[k+1][c]} is one contiguous 8-byte ds_load_b64.
template<bool HAS_ADD, bool HAS_BIAS, bool RELU>
__global__ __launch_bounds__(256)
void gemm128_wmma(const float* __restrict__ Xa,
                  const float* __restrict__ Xb,
                  const float* __restrict__ W,
                  const float* __restrict__ bias,
                  float* __restrict__ out, int M)
{
    __shared__ float sW[HID * HID];   // 64 KB

    const int tid = threadIdx.x;
    // Cooperative pair-interleave load: 64 kpairs x 64 col-pairs = 4096 float4 stores
    #pragma unroll
    for (int it = 0; it < 16; ++it) {
        const int i  = tid + it * 256;    // 0..4095
        const int kp = i >> 6;            // 0..63
        const int c2 = (i & 63) * 2;      // even column
        const float2 r0 = *(const float2*)(W + (size_t)(2 * kp)     * HID + c2);
        const float2 r1 = *(const float2*)(W + (size_t)(2 * kp + 1) * HID + c2);
        float4 v; v.x = r0.x; v.y = r1.x; v.z = r0.y; v.w = r1.y;
        *(float4*)(sW + kp * 256 + c2 * 2) = v;
    }
    __syncthreads();

    const int wave  = tid >> 5;
    const int lane  = tid & 31;
    const int mtile = blockIdx.x * 8 + wave;      // one 16-row tile per wave
    if (mtile * 16 >= M) return;                  // wave-uniform exit (EXEC all-1 for WMMA)
    const int row0  = mtile * 16;

    const int mrow  = lane & 15;   // A: M index; B: N index; C/D: column index
    const int khalf = lane >> 4;   // selects K offset {0,2} within each K-group of 4

    const float* __restrict__ Apos = Xa + (size_t)(row0 + mrow) * HID + 2 * khalf;
    const float* __restrict__ Xpos = HAS_ADD ? (Xb + (size_t)(row0 + mrow) * HID + 2 * khalf)
                                             : nullptr;
    // B fragment base: kpair = k0/2 + khalf, col = n*16 + mrow
    // addr(float) = (k0/2 + khalf)*256 + (n*16 + mrow)*2 = [khalf*256 + mrow*2] + k0*128 + n*32
    const float* __restrict__ Bbase = sW + khalf * 256 + mrow * 2;

    v8f acc[8] = {};   // 16 x 128 f32 output strip, 8 N-tiles

    for (int k0 = 0; k0 < HID; k0 += 4) {
        // A fragment: lane holds (M=mrow, K = k0 + 2*khalf + {0,1})
        float2 av = *(const float2*)(Apos + k0);
        if (HAS_ADD) {
            float2 xv = *(const float2*)(Xpos + k0);
            av.x += xv.x; av.y += xv.y;
        }
        v2f a; a.x = av.x; a.y = av.y;

        #pragma unroll
        for (int n = 0; n < 8; ++n) {
            const v2f b = *(const v2f*)(Bbase + k0 * 128 + n * 32);  // ds_load_b64, imm offset
            acc[n] = __builtin_amdgcn_wmma_f32_16x16x4_f32(
                false, a, false, b, (short)0, acc[n], false, false);
        }
    }

    // C/D layout: reg r, lane l -> (M = r + 8*(l>>4), column = l&15 within N-tile)
    #pragma unroll
    for (int n = 0; n < 8; ++n) {
        float bval = 0.0f;
        if (HAS_BIAS) bval = bias[n * 16 + mrow];
        float* __restrict__ ocol = out + (size_t)row0 * HID + n * 16 + mrow;
        #pragma unroll
        for (int r = 0; r < 8; ++r) {
            const int m = r + 8 * khalf;
            float v = acc[n][r] + bval;
            if (RELU) v = fmaxf(v, 0.0f);
            ocol[(size_t)m * HID] = v;
        }
    }
}

// One wave per edge: gather h[src] row (float4 per lane), atomic-add into agg[dst].
// h (51.2 MB) is L2-resident on MI455X (192 MB L2) -> gathers + atomics resolve in L2.
__global__ __launch_bounds__(256)
void scatter_edges(const int* __restrict__ src, const int* __restrict__ dst,
                   const float* __restrict__ h, float* __restrict__ agg, int nEdges)
{
    const int t    = blockIdx.x * 256 + threadIdx.x;
    const int e    = t >> 5;
    const int lane = t & 31;
    if (e >= nEdges) return;
    const int s = src[e];
    const int d = dst[e];
    const float4 v = *(const float4*)(h + (size_t)s * HID + lane * 4);
    float* __restrict__ p = agg + (size_t)d * HID + lane * 4;
    atomicAdd(p + 0, v.x);
    atomicAdd(p + 1, v.y);
    atomicAdd(p + 2, v.z);
    atomicAdd(p + 3, v.w);
}

// One wave per node: global add-pool into out[batch[node]]
__global__ __launch_bounds__(256)
void pool_kernel(const float* __restrict__ h, const int* __restrict__ batch,
                 float* __restrict__ out, int nNodes)
{
    const int t    = blockIdx.x * 256 + threadIdx.x;
    const int node = t >> 5;
    const int lane = t & 31;
    if (node >= nNodes) return;
    const int g = batch[node];
    const float4 v = *(const float4*)(h + (size_t)node * HID + lane * 4);
    float* __restrict__ p = out + (size_t)g * HID + lane * 4;
    atomicAdd(p + 0, v.x);
    atomicAdd(p + 1, v.y);
    atomicAdd(p + 2, v.z);
    atomicAdd(p + 3, v.w);
}

extern "C" void kernel_launch(void* const* d_in, const int* in_sizes, int n_in,
                              void* d_out, int out_size, void* d_ws, size_t ws_size,
                              hipStream_t stream)
{
    (void)in_sizes; (void)n_in; (void)out_size; (void)ws_size;

    const float* x     = (const float*)d_in[0];
    const int*   ei    = (const int*)  d_in[1];   // [2, NE]: src row then dst row
    const int*   batch = (const int*)  d_in[2];
    const float* Wproj = (const float*)d_in[3];
    const float* W1    = (const float*)d_in[4];
    const float* b1    = (const float*)d_in[5];
    const float* W2    = (const float*)d_in[6];
    const float* b2    = (const float*)d_in[7];
    float*       out   = (float*)d_out;

    float* h   = (float*)d_ws;                    // [NN,128] current features
    float* agg = h + (size_t)NN * HID;            // [NN,128] aggregation / z (aliased)

    const int nMTiles    = (NN + 15) / 16;                 // 6250
    const int gemmBlocks = (nMTiles + 7) / 8;              // 782
    const int scatBlocks = (int)(((long)NE * 32 + 255) / 256);
    const int poolBlocks = (int)(((long)NN * 32 + 255) / 256);

    // h = x @ Wproj
    gemm128_wmma<false, false, false><<<gemmBlocks, 256, 0, stream>>>(
        x, nullptr, Wproj, nullptr, h, NN);

    for (int l = 0; l < NLAY; ++l) {
        hipMemsetAsync(agg, 0, (size_t)NN * HID * sizeof(float), stream);
        scatter_edges<<<scatBlocks, 256, 0, stream>>>(ei, ei + NE, h, agg, NE);
        // z = relu((h + agg) @ W1[l] + b1[l]); z written in-place over agg (safe per-wave)
        gemm128_wmma<true, true, true><<<gemmBlocks, 256, 0, stream>>>(
            h, agg, W1 + (size_t)l * HID * HID, b1 + l * HID, agg, NN);
        // h = relu(z @ W2[l] + b2[l])
        gemm128_wmma<false, true, true><<<gemmBlocks, 256, 0, stream>>>(
            agg, nullptr, W2 + (size_t)l * HID * HID, b2 + l * HID, h, NN);
    }

    hipMemsetAsync(out, 0, (size_t)NG * HID * sizeof(float), stream);
    pool_kernel<<<poolBlocks, 256, 0, stream>>>(h, batch, out, NN);
}